// UnitSphereNorm_39556648796522
// MI455X (gfx1250) — compile-verified
//
#include <hip/hip_runtime.h>
#include <hip/hip_bf16.h>

#define TILE 1024          // points per async tile (12 KB of pos data)
#define NBLK 2048          // blocks for the streaming normalize pass

// ---------- order-preserving float <-> uint mapping (for atomic min/max) ----
__device__ __forceinline__ unsigned f2u(float f) {
  unsigned u = __float_as_uint(f);
  return (u & 0x80000000u) ? ~u : (u | 0x80000000u);
}
__device__ __forceinline__ float u2f(unsigned u) {
  unsigned v = (u & 0x80000000u) ? (u & 0x7FFFFFFFu) : ~u;
  return __uint_as_float(v);
}

// ---------- CDNA5 async global->LDS helpers --------------------------------
__device__ __forceinline__ void async_b128(unsigned lds_addr, unsigned long long gaddr) {
  // GFX1250 VFLAT async op, tracked with ASYNCcnt (CDNA5 ISA)
  asm volatile("global_load_async_to_lds_b128 %0, %1, off"
               :: "v"(lds_addr), "v"(gaddr) : "memory");
}
template <int CNT>
__device__ __forceinline__ void wait_asynccnt() {
#if __has_builtin(__builtin_amdgcn_s_wait_asynccnt)
  __builtin_amdgcn_s_wait_asynccnt(CNT);
#else
  asm volatile("s_wait_asynccnt %0" :: "i"(CNT) : "memory");
#endif
}

// copy TILE points (TILE*12 bytes) into LDS: 3 x b128 per thread (256 threads)
__device__ __forceinline__ void async_tile(const float* __restrict__ gsrc,
                                           unsigned lds_base, int tid) {
  const char* base = (const char*)gsrc;
#pragma unroll
  for (int k = 0; k < 3; ++k) {
    unsigned off = (unsigned)(tid + k * 256) * 16u;
    async_b128(lds_base + off, (unsigned long long)(base + off));
  }
}

// ---------- kernel 0: initialize workspace accumulators --------------------
__global__ __launch_bounds__(256) void init_ws_kernel(unsigned* __restrict__ ws, int S) {
  int i = blockIdx.x * blockDim.x + threadIdx.x;
  int total = 10 * S;
  if (i >= total) return;
  unsigned v;
  if (i < 3 * S)       v = 0xFF800000u;  // f2u(+inf): min identity
  else if (i < 6 * S)  v = 0x007FFFFFu;  // f2u(-inf): max identity
  else                 v = 0u;           // 0.0f for the sums
  ws[i] = v;
}

// ---------- kernel 1: segmented min/max + weighted sums --------------------
__global__ __launch_bounds__(256) void seg_stats_kernel(
    const float* __restrict__ pos, const int* __restrict__ idx,
    const float* __restrict__ w, unsigned* __restrict__ mnk,
    unsigned* __restrict__ mxk, float* __restrict__ wsum,
    float* __restrict__ wps, int N, int S) {
  int i = blockIdx.x * blockDim.x + threadIdx.x;
  bool valid = (i < N);
  int seg = -1;
  float x = 0.f, y = 0.f, z = 0.f, wi = 0.f;
  if (valid) {
    seg = idx[i];                               // re-used in pass 3: keep in L2 (RT)
    x = pos[3 * i + 0]; y = pos[3 * i + 1]; z = pos[3 * i + 2];  // re-used: RT
    wi = __builtin_nontemporal_load(&w[i]);     // read-once: NT, don't pollute L2
  }
  unsigned kx = f2u(x), ky = f2u(y), kz = f2u(z);
  float wx = wi * x, wy = wi * y, wz = wi * z;

  int seg0 = __shfl(seg, 0, 32);
  bool uniform = __all(seg == seg0) && (seg0 >= 0);
  if (uniform) {
    // whole wave in one segment (common: ~256 pts/segment, sorted idx)
    unsigned ax = kx, ay = ky, az = kz;   // mins
    unsigned bx = kx, by = ky, bz = kz;   // maxs
    float sw = wi, sx = wx, sy = wy, sz = wz;
#pragma unroll
    for (int off = 16; off > 0; off >>= 1) {
      ax = min(ax, __shfl_xor(ax, off, 32));
      ay = min(ay, __shfl_xor(ay, off, 32));
      az = min(az, __shfl_xor(az, off, 32));
      bx = max(bx, __shfl_xor(bx, off, 32));
      by = max(by, __shfl_xor(by, off, 32));
      bz = max(bz, __shfl_xor(bz, off, 32));
      sw += __shfl_xor(sw, off, 32);
      sx += __shfl_xor(sx, off, 32);
      sy += __shfl_xor(sy, off, 32);
      sz += __shfl_xor(sz, off, 32);
    }
    if ((threadIdx.x & 31) == 0) {
      atomicMin(&mnk[0 * S + seg0], ax);
      atomicMin(&mnk[1 * S + seg0], ay);
      atomicMin(&mnk[2 * S + seg0], az);
      atomicMax(&mxk[0 * S + seg0], bx);
      atomicMax(&mxk[1 * S + seg0], by);
      atomicMax(&mxk[2 * S + seg0], bz);
      atomicAdd(&wsum[seg0], sw);
      atomicAdd(&wps[0 * S + seg0], sx);
      atomicAdd(&wps[1 * S + seg0], sy);
      atomicAdd(&wps[2 * S + seg0], sz);
    }
  } else if (valid) {
    // segment-boundary wave: per-lane atomics (rare: ~12% of waves, 1 boundary each)
    atomicMin(&mnk[0 * S + seg], kx);
    atomicMin(&mnk[1 * S + seg], ky);
    atomicMin(&mnk[2 * S + seg], kz);
    atomicMax(&mxk[0 * S + seg], kx);
    atomicMax(&mxk[1 * S + seg], ky);
    atomicMax(&mxk[2 * S + seg], kz);
    atomicAdd(&wsum[seg], wi);
    atomicAdd(&wps[0 * S + seg], wx);
    atomicAdd(&wps[1 * S + seg], wy);
    atomicAdd(&wps[2 * S + seg], wz);
  }
}

// ---------- kernel 2: finalize per-segment center / diameter ---------------
__global__ __launch_bounds__(256) void seg_finalize_kernel(
    const unsigned* __restrict__ mnk, const unsigned* __restrict__ mxk,
    const float* __restrict__ wsum, const float* __restrict__ wps,
    float* __restrict__ cx, float* __restrict__ cy, float* __restrict__ cz,
    float* __restrict__ denom, float* __restrict__ out_diam, int S) {
  int s = blockIdx.x * blockDim.x + threadIdx.x;
  if (s >= S) return;
  float ex = u2f(mxk[0 * S + s]) - u2f(mnk[0 * S + s]);
  float ey = u2f(mxk[1 * S + s]) - u2f(mnk[1 * S + s]);
  float ez = u2f(mxk[2 * S + s]) - u2f(mnk[2 * S + s]);
  float diam = fmaxf(fmaxf(ex, ey), ez);
  float sw = wsum[s];
  float safe = (sw == 0.0f) ? 1.0f : sw;
  cx[s] = wps[0 * S + s] / safe;
  cy[s] = wps[1 * S + s] / safe;
  cz[s] = wps[2 * S + s] / safe;
  denom[s] = diam + 0.01f;
  out_diam[s] = diam;
}

// ---------- kernel 3: normalize points, pos staged via async LDS pipeline --
__global__ __launch_bounds__(256) void normalize_kernel(
    const float* __restrict__ pos, const int* __restrict__ idx,
    const float* __restrict__ cx, const float* __restrict__ cy,
    const float* __restrict__ cz, const float* __restrict__ denom,
    float* __restrict__ out, int ntiles) {
  __shared__ __align__(16) float buf[2][TILE * 3];
  const int tid = threadIdx.x;
  const int step = gridDim.x;
  unsigned lds0 = (unsigned)(unsigned long long)(&buf[0][0]);
  unsigned lds1 = (unsigned)(unsigned long long)(&buf[1][0]);

  int t = blockIdx.x;
  if (t < ntiles) async_tile(pos + (size_t)t * TILE * 3, lds0, tid);
  int p = 0;
  for (; t < ntiles; t += step) {
    int tn = t + step;
    if (tn < ntiles) {
      async_tile(pos + (size_t)tn * TILE * 3, p ? lds0 : lds1, tid);
      wait_asynccnt<3>();   // current tile's 3 copies done; next tile in flight
    } else {
      wait_asynccnt<0>();
    }
    __syncthreads();
    const float* b = buf[p];
#pragma unroll
    for (int k = 0; k < TILE / 256; ++k) {
      int j = k * 256 + tid;
      long long g = (long long)t * TILE + j;
      int seg = idx[g];                 // 2nd use of idx: should hit L2
      float d = denom[seg];
      float ox = (b[3 * j + 0] - cx[seg]) / d;
      float oy = (b[3 * j + 1] - cy[seg]) / d;
      float oz = (b[3 * j + 2] - cz[seg]) / d;
      // write-once 96 MB stream: NT stores so write-allocate doesn't evict
      // the pos/idx working set (128 MB fits the 192 MB L2)
      __builtin_nontemporal_store(ox, &out[3 * g + 0]);
      __builtin_nontemporal_store(oy, &out[3 * g + 1]);
      __builtin_nontemporal_store(oz, &out[3 * g + 2]);
    }
    __syncthreads();  // protect buf[p] before it is re-targeted by async copy
    p ^= 1;
  }
}

// ---------- tail (only if N not a multiple of TILE; here N = 2^23 so unused)
__global__ __launch_bounds__(256) void normalize_tail_kernel(
    const float* __restrict__ pos, const int* __restrict__ idx,
    const float* __restrict__ cx, const float* __restrict__ cy,
    const float* __restrict__ cz, const float* __restrict__ denom,
    float* __restrict__ out, int start, int N) {
  int i = start + blockIdx.x * blockDim.x + threadIdx.x;
  if (i >= N) return;
  int seg = idx[i];
  float d = denom[seg];
  __builtin_nontemporal_store((pos[3 * i + 0] - cx[seg]) / d, &out[3 * i + 0]);
  __builtin_nontemporal_store((pos[3 * i + 1] - cy[seg]) / d, &out[3 * i + 1]);
  __builtin_nontemporal_store((pos[3 * i + 2] - cz[seg]) / d, &out[3 * i + 2]);
}

extern "C" void kernel_launch(void* const* d_in, const int* in_sizes, int n_in,
                              void* d_out, int out_size, void* d_ws, size_t ws_size,
                              hipStream_t stream) {
  const float* pos = (const float*)d_in[0];
  const int*   idx = (const int*)d_in[1];
  const float* w   = (const float*)d_in[2];
  // d_in[3] = num_super (device scalar, unused: derive S from sizes)

  const int N = in_sizes[0] / 3;          // pos is [N,3]
  const int S = out_size - in_sizes[0];   // out = N*3 pos_out + S diameters

  float* out_pos  = (float*)d_out;
  float* out_diam = out_pos + (size_t)N * 3;

  // workspace layout (32-bit words): 6S min/max keys | S wsum | 3S wpos | 4S finals
  unsigned* ws   = (unsigned*)d_ws;
  unsigned* mnk  = ws;
  unsigned* mxk  = ws + 3 * (size_t)S;
  float* wsum    = (float*)(ws + 6 * (size_t)S);
  float* wps     = (float*)(ws + 7 * (size_t)S);
  float* cx      = (float*)(ws + 10 * (size_t)S);
  float* cy      = (float*)(ws + 11 * (size_t)S);
  float* cz      = (float*)(ws + 12 * (size_t)S);
  float* denom   = (float*)(ws + 13 * (size_t)S);

  int initWords = 10 * S;
  init_ws_kernel<<<(initWords + 255) / 256, 256, 0, stream>>>(ws, S);

  seg_stats_kernel<<<(N + 255) / 256, 256, 0, stream>>>(pos, idx, w, mnk, mxk,
                                                        wsum, wps, N, S);

  seg_finalize_kernel<<<(S + 255) / 256, 256, 0, stream>>>(mnk, mxk, wsum, wps,
                                                           cx, cy, cz, denom,
                                                           out_diam, S);

  int ntiles = N / TILE;
  if (ntiles > 0) {
    int grid = ntiles < NBLK ? ntiles : NBLK;
    normalize_kernel<<<grid, 256, 0, stream>>>(pos, idx, cx, cy, cz, denom,
                                               out_pos, ntiles);
  }
  int rem = N - ntiles * TILE;
  if (rem > 0) {
    normalize_tail_kernel<<<(rem + 255) / 256, 256, 0, stream>>>(
        pos, idx, cx, cy, cz, denom, out_pos, ntiles * TILE, N);
  }
}